// yolov8_target_22273700397212
// MI455X (gfx1250) — compile-verified
//
#include <hip/hip_runtime.h>
#include <stdint.h>

#define CONF 2.5f
#define NCLS 80
#define TILE_ROWS 128
#define THREADS 128
#define ROW_PAD 84          // floats per LDS row: 80 data + 4 pad -> 336B stride, conflict-free b128 reads
#define CHUNKS_PER_ROW 20   // 80 floats = 20 x float4
#define TILE_CHUNKS (TILE_ROWS * CHUNKS_PER_ROW)

// Issue one CDNA5 async global->LDS 16B copy. lds_addr is the wave-relative
// LDS byte offset (low 32 bits of a generic LDS pointer, ISA 10.2).
__device__ __forceinline__ void async_copy16(uint32_t lds_addr, uint64_t gaddr) {
    asm volatile("global_load_async_to_lds_b128 %0, %1, off"
                 :: "v"(lds_addr), "v"(gaddr)
                 : "memory");
}

__global__ __launch_bounds__(THREADS)
void yolo_filter_partial(const float* __restrict__ post,
                         const float* __restrict__ boxes,
                         float* __restrict__ partial, int n) {
    __shared__ __align__(16) float tile[TILE_ROWS * ROW_PAD];
    __shared__ float red[THREADS / 32];

    const int tid = (int)threadIdx.x;
    const int base = (int)blockIdx.x * TILE_ROWS;
    const int valid_rows = min(TILE_ROWS, n - base);
    const int valid_chunks = valid_rows * CHUNKS_PER_ROW;

    const uint32_t lds_base = (uint32_t)(uintptr_t)(&tile[0]);
    const uint64_t gbase = (uint64_t)(uintptr_t)post + (uint64_t)base * (NCLS * 4);

    // ---- Stage the 128x320B score tile into LDS (linear global stream,
    // row-padded LDS layout: lds_off = i*16 + (i/20)*16).
    if (valid_chunks == TILE_CHUNKS) {
        // Fast path (2343 of 2344 blocks): no guards, straight clause of
        // 20 async loads per thread, adjacent lanes -> adjacent 16B.
#pragma unroll
        for (int k = 0; k < TILE_CHUNKS / THREADS; ++k) {
            uint32_t i = (uint32_t)(tid + k * THREADS);
            uint32_t row = i / CHUNKS_PER_ROW;
            async_copy16(lds_base + i * 16u + row * 16u,
                         gbase + (uint64_t)i * 16u);
        }
    } else {
        // Tail block: branchless clamp (duplicate writes of the last valid chunk are
        // harmless: same source -> same destination).
#pragma unroll
        for (int k = 0; k < TILE_CHUNKS / THREADS; ++k) {
            int iraw = tid + k * THREADS;
            uint32_t i = (uint32_t)min(iraw, valid_chunks - 1);
            uint32_t row = i / CHUNKS_PER_ROW;
            async_copy16(lds_base + i * 16u + row * 16u,
                         gbase + (uint64_t)i * 16u);
        }
    }
    // Each wave waits for its own async copies; the barrier publishes all
    // waves' LDS writes to the workgroup.
    asm volatile("s_wait_asynccnt 0" ::: "memory");
    __syncthreads();

    // ---- Per-row max over 80 classes, threshold, add box coords.
    float contrib = 0.0f;
    if (tid < valid_rows) {
        const float4* rowp = (const float4*)&tile[tid * ROW_PAD]; // 336B-aligned
        float m = -3.402823466e38f;
#pragma unroll
        for (int j = 0; j < CHUNKS_PER_ROW; ++j) {
            float4 v = rowp[j];            // ds_load_b128, conflict-free (stride 84 dwords)
            m = fmaxf(m, fmaxf(fmaxf(v.x, v.y), fmaxf(v.z, v.w)));
        }
        if (m >= CONF) {
            float4 b = ((const float4*)boxes)[base + tid]; // coalesced global_load_b128
            contrib = m + ((b.x + b.y) + (b.z + b.w));
        }
    }

    // ---- Deterministic reduction: wave-level shuffle tree (no barriers),
    // then one LDS combine across the 4 waves.
#pragma unroll
    for (int off = 16; off > 0; off >>= 1)
        contrib += __shfl_xor(contrib, off, 32);

    const int wave = tid >> 5;
    const int lane = tid & 31;
    if (lane == 0) red[wave] = contrib;
    __syncthreads();
    if (tid == 0) {
        float s = red[0] + red[1] + red[2] + red[3];
        partial[blockIdx.x] = s;
    }
}

__global__ __launch_bounds__(256)
void yolo_reduce_final(const float* __restrict__ partial, int nblk,
                       float* __restrict__ out) {
    __shared__ float red[256 / 32];
    float s = 0.0f;
    for (int i = (int)threadIdx.x; i < nblk; i += 256) s += partial[i];
#pragma unroll
    for (int off = 16; off > 0; off >>= 1)
        s += __shfl_xor(s, off, 32);
    const int wave = (int)threadIdx.x >> 5;
    const int lane = (int)threadIdx.x & 31;
    if (lane == 0) red[wave] = s;
    __syncthreads();
    if (threadIdx.x == 0) {
        float t = 0.0f;
#pragma unroll
        for (int w = 0; w < 256 / 32; ++w) t += red[w];
        out[0] = t;
    }
}

extern "C" void kernel_launch(void* const* d_in, const int* in_sizes, int n_in,
                              void* d_out, int out_size, void* d_ws, size_t ws_size,
                              hipStream_t stream) {
    const float* post  = (const float*)d_in[0];  // [N, 80] f32
    const float* boxes = (const float*)d_in[1];  // [N, 4]  f32
    float* out     = (float*)d_out;              // scalar f32
    float* partial = (float*)d_ws;               // nblk floats of scratch

    const int n    = in_sizes[0] / NCLS;         // 300000
    const int nblk = (n + TILE_ROWS - 1) / TILE_ROWS;

    yolo_filter_partial<<<nblk, THREADS, 0, stream>>>(post, boxes, partial, n);
    yolo_reduce_final<<<1, 256, 0, stream>>>(partial, nblk, out);
}